// GIN_1082331759084
// MI455X (gfx1250) — compile-verified
//
#include <hip/hip_runtime.h>

#define N_NODES 50000
#define N_EDGES 800000
#define D_IN    64
#define D_HID   128
#define D_OUT   128

typedef __attribute__((ext_vector_type(2))) float v2f;
typedef __attribute__((ext_vector_type(8))) float v8f;

// ---------------------------------------------------------------------------
// agg = src buffer copy (float4-vectorized)
// ---------------------------------------------------------------------------
__global__ void gin_copy_f4_kernel(const float4* __restrict__ in,
                                   float4* __restrict__ out, int n4) {
  int i = blockIdx.x * blockDim.x + threadIdx.x;
  if (i < n4) out[i] = in[i];
}

// ---------------------------------------------------------------------------
// Edge-parallel scatter-add: AGG[dst] += H[src], D floats per edge.
// One thread per float4 chunk -> 4 hardware f32 atomics. agg is L2-resident.
// ---------------------------------------------------------------------------
template <int D>
__global__ void gin_scatter_add_kernel(const long long* __restrict__ src,
                                       const long long* __restrict__ dst,
                                       const float* __restrict__ H,
                                       float* __restrict__ AGG) {
  constexpr int CH = D / 4;  // float4 chunks per edge (16 or 32)
  long long t = (long long)blockIdx.x * blockDim.x + threadIdx.x;
  long long e = t / CH;      // CH is a power of two -> shift
  int c = (int)(t % CH);
  if (e >= N_EDGES) return;
  long long s = src[e];
  long long d = dst[e];
  float4 v = ((const float4*)(H + s * D))[c];
  float* ap = AGG + d * D + c * 4;
  unsafeAtomicAdd(ap + 0, v.x);
  unsafeAtomicAdd(ap + 1, v.y);
  unsafeAtomicAdd(ap + 2, v.z);
  unsafeAtomicAdd(ap + 3, v.w);
}

// ---------------------------------------------------------------------------
// Y[16 x 128 tile] = relu(A[16 x K] @ W[K x 128] + bias) via
// V_WMMA_F32_16X16X4_F32 (fp32 in / fp32 accumulate).
// Block = 256 threads = 8 waves; wave w owns N-tile w; block owns one M-tile.
// A tile staged in LDS (padded stride to avoid bank conflicts); W/bias stream
// from L2 (W is <=64 KB, fully resident).
// ---------------------------------------------------------------------------
template <int K>
__global__ __launch_bounds__(256)
void gin_gemm_relu_kernel(const float* __restrict__ A,
                          const float* __restrict__ W,     // [K][128]
                          const float* __restrict__ bias,  // [128]
                          float* __restrict__ Y) {         // [M][128]
  constexpr int AP = K + 4;          // padded LDS row stride (floats)
  __shared__ float sA[16 * AP];
  __shared__ float sB[128];

  const int tid = threadIdx.x;
  const int mtile = blockIdx.x;

  // Cooperative stage of the 16 x K A-tile into LDS.
  const float* Ab = A + (size_t)mtile * 16 * K;
  for (int i = tid; i < 16 * K; i += 256) {
    int r = i / K;
    int c = i % K;
    sA[r * AP + c] = Ab[i];
  }
  if (tid < 128) sB[tid] = bias[tid];
  __syncthreads();

  const int lane = tid & 31;
  const int nt   = tid >> 5;        // N-tile index 0..7
  const int lrow = lane & 15;       // M (for A) / N (for B,C) within tile
  const int kh   = (lane >> 4) << 1;  // half-wave K offset: 0 or 2

  const float* sArow = sA + lrow * AP;
  const float* Wc    = W + nt * 16 + lrow;   // column (nt*16 + lrow)

  v8f acc = {};
  for (int k0 = 0; k0 < K; k0 += 4) {
    v2f a, b;
    // A 16x4 f32 fragment: lane holds (M=lrow, K=k0+kh) and (M=lrow, K=k0+kh+1)
    a.x = sArow[k0 + kh + 0];
    a.y = sArow[k0 + kh + 1];
    // B 4x16 f32 fragment: lane holds (K=k0+kh, N) and (K=k0+kh+1, N)
    b.x = Wc[(size_t)(k0 + kh + 0) * 128];
    b.y = Wc[(size_t)(k0 + kh + 1) * 128];
    acc = __builtin_amdgcn_wmma_f32_16x16x4_f32(
        /*neg_a=*/false, a, /*neg_b=*/false, b,
        /*c_mod=*/(short)0, acc, /*reuse_a=*/false, /*reuse_b=*/false);
  }

  // Epilogue: C/D layout -> VGPR r holds row M = r + 8*(lane>=16), col N = lrow.
  const float bn = sB[nt * 16 + lrow];
  float* Yp = Y + ((size_t)mtile * 16 + ((lane >> 4) << 3)) * 128 + nt * 16 + lrow;
#pragma unroll
  for (int r = 0; r < 8; ++r) {
    float v = acc[r] + bn;
    Yp[(size_t)r * 128] = fmaxf(v, 0.0f);
  }
}

// ---------------------------------------------------------------------------
extern "C" void kernel_launch(void* const* d_in, const int* in_sizes, int n_in,
                              void* d_out, int out_size, void* d_ws, size_t ws_size,
                              hipStream_t stream) {
  (void)in_sizes; (void)n_in; (void)out_size; (void)ws_size;

  const long long* edge = (const long long*)d_in[0];  // int64 [2][E]
  const long long* src  = edge;                        // row 0: x_j source
  const long long* dst  = edge + N_EDGES;              // row 1: target node i
  const float* x  = (const float*)d_in[1];
  const float* w1 = (const float*)d_in[2];
  const float* b1 = (const float*)d_in[3];
  const float* w2 = (const float*)d_in[4];
  const float* b2 = (const float*)d_in[5];
  float* out = (float*)d_out;

  // Workspace layout (fully rewritten each call; 64 MB total)
  float* agg1 = (float*)d_ws;                          // N * 64
  float* h    = agg1 + (size_t)N_NODES * D_IN;         // N * 128
  float* agg2 = h + (size_t)N_NODES * D_HID;           // N * 128

  // ---- Layer 1: h = relu((x + scatter(x)) @ w1 + b1) ----
  {
    int n4 = N_NODES * D_IN / 4;
    gin_copy_f4_kernel<<<(n4 + 255) / 256, 256, 0, stream>>>(
        (const float4*)x, (float4*)agg1, n4);
    long long tot = (long long)N_EDGES * (D_IN / 4);
    gin_scatter_add_kernel<D_IN><<<(int)((tot + 255) / 256), 256, 0, stream>>>(
        src, dst, x, agg1);
    gin_gemm_relu_kernel<D_IN><<<N_NODES / 16, 256, 0, stream>>>(
        agg1, w1, b1, h);
  }

  // ---- Layer 2: out = relu((h + scatter(h)) @ w2 + b2) ----
  {
    int n4 = N_NODES * D_HID / 4;
    gin_copy_f4_kernel<<<(n4 + 255) / 256, 256, 0, stream>>>(
        (const float4*)h, (float4*)agg2, n4);
    long long tot = (long long)N_EDGES * (D_HID / 4);
    gin_scatter_add_kernel<D_HID><<<(int)((tot + 255) / 256), 256, 0, stream>>>(
        src, dst, h, agg2);
    gin_gemm_relu_kernel<D_HID><<<N_NODES / 16, 256, 0, stream>>>(
        agg2, w2, b2, out);
  }
}